// GAT_47442208751847
// MI455X (gfx1250) — compile-verified
//
#include <hip/hip_runtime.h>
#include <hip/hip_bf16.h>

#define N_NODES 50000
#define N_EDGES 800000
#define IN_DIM  128
#define HID     32
#define HEADS   4
#define OUT_DIM 64
#define NEG_SLOPE 0.2f

typedef __attribute__((ext_vector_type(2))) float v2f;
typedef __attribute__((ext_vector_type(8))) float v8f;
typedef __attribute__((ext_vector_type(4))) unsigned int u32x4;
typedef __attribute__((ext_vector_type(8))) int i32x8;
typedef __attribute__((ext_vector_type(4))) int i32x4;

// ---------------------------------------------------------------------------
// fp32 GEMM: TDM-staged B panel in LDS + V_WMMA_F32_16X16X4_F32.
// A: [M x K] row-major (global), B: [K x N] row-major (DMA'd to LDS once per
// block by the Tensor Data Mover), C: [M x N] row-major.
// grid.x = M/16, blockDim.x = 32 * (N/16); one wave per 16x16 output tile.
// ---------------------------------------------------------------------------
template <int K, int NCOLS>
__global__ void gat_gemm_wmma_tdm(const float* __restrict__ A,
                                  const float* __restrict__ B,
                                  float* __restrict__ C) {
  __shared__ float Bs[K * NCOLS];

  // --- wave 0: issue one TDM load of the whole B panel (K x NCOLS f32) ---
  if (threadIdx.x < 32) {
    unsigned long long gaddr = (unsigned long long)(uintptr_t)B;
    unsigned int ldsoff = (unsigned int)(uintptr_t)Bs;  // flat addr low 32b = LDS byte offset

    u32x4 g0;
    g0.x = 1u;                                        // count=1, user descriptor
    g0.y = ldsoff;                                    // lds_addr (bytes)
    g0.z = (unsigned int)(gaddr & 0xffffffffu);       // global_addr[31:0]
    g0.w = (unsigned int)((gaddr >> 32) & 0x01ffffffu)// global_addr[56:32]
         | (2u << 30);                                // type = 2 ("image")

    i32x8 g1;
    g1[0] = (2 << 16);                                // data_size=2 (4B); mask=0; no flags
    g1[1] = (NCOLS & 0xffff) << 16;                   // tensor_dim0[15:0]
    g1[2] = ((NCOLS >> 16) & 0xffff)                  // tensor_dim0[31:16]
          | ((K & 0xffff) << 16);                     // tensor_dim1[15:0]
    g1[3] = ((K >> 16) & 0xffff)                      // tensor_dim1[31:16]
          | ((NCOLS & 0xffff) << 16);                 // tile_dim0 = NCOLS
    g1[4] = (K & 0xffff);                             // tile_dim1 = K rows; tile_dim2=0
    g1[5] = NCOLS;                                    // tensor_dim0_stride[31:0]
    g1[6] = 0;                                        // stride hi / dim1_stride lo (2D: unused)
    g1[7] = 0;

    i32x4 gz4 = {0, 0, 0, 0};                         // groups 2/3 unused (<=2D tensor)
    i32x8 gz8 = {0, 0, 0, 0, 0, 0, 0, 0};
    __builtin_amdgcn_tensor_load_to_lds(g0, g1, gz4, gz4, gz8, /*cpol=*/0);
    __builtin_amdgcn_s_wait_tensorcnt(0);
  }
  __syncthreads();

  const int lane  = threadIdx.x & 31;
  const int wave  = threadIdx.x >> 5;
  const int m0    = blockIdx.x << 4;
  const int n0    = wave << 4;
  const int l15   = lane & 15;
  const int khalf = (lane >> 4) << 1;

  const float* Arow = A + (long)(m0 + l15) * K;
  const float* Bcol = Bs + n0 + l15;

  v8f acc = {};
#pragma unroll 4
  for (int k = 0; k < K; k += 4) {
    const int ka = k + khalf;
    v2f a = *(const v2f*)(Arow + ka);                 // aligned: ka even, K even
    v2f b;
    b.x = Bcol[ka * NCOLS];                           // ds_load from staged panel
    b.y = Bcol[(ka + 1) * NCOLS];
    acc = __builtin_amdgcn_wmma_f32_16x16x4_f32(
        /*neg_a=*/false, a, /*neg_b=*/false, b,
        /*c_mod=*/(short)0, acc, /*reuse_a=*/false, /*reuse_b=*/false);
  }

  const int mbase = m0 + ((lane >> 4) << 3);
  float* Cp = C + (long)mbase * NCOLS + n0 + l15;
#pragma unroll
  for (int r = 0; r < 8; ++r) Cp[(long)r * NCOLS] = acc[r];
}

// ---------------------------------------------------------------------------
// Per-(node, head) attention coefficients: dot(h[n,h,:], a_src/a_dst[h,:])
// ---------------------------------------------------------------------------
__global__ void gat_attn_coef(const float* __restrict__ h,
                              const float* __restrict__ a_src,
                              const float* __restrict__ a_dst,
                              float* __restrict__ as_out,
                              float* __restrict__ ad_out,
                              int n_nodes, int heads, int F) {
  int idx = blockIdx.x * blockDim.x + threadIdx.x;
  if (idx >= n_nodes * heads) return;
  int node = idx / heads;
  int hd   = idx - node * heads;
  const float* hp = h + (long)node * heads * F + (long)hd * F;
  const float* sp = a_src + (long)hd * F;
  const float* dp = a_dst + (long)hd * F;
  float s = 0.f, d = 0.f;
  for (int f = 0; f < F; ++f) {
    float v = hp[f];
    s += v * sp[f];
    d += v * dp[f];
  }
  as_out[idx] = s;
  ad_out[idx] = d;
}

__device__ __forceinline__ float gat_leaky(float v) {
  return v >= 0.f ? v : NEG_SLOPE * v;
}
// order-preserving float<->int encoding for integer atomicMax
__device__ __forceinline__ int gat_f2ord(float f) {
  int i = __float_as_int(f);
  return i >= 0 ? i : (i ^ 0x7fffffff);
}
__device__ __forceinline__ float gat_ord2f(int i) {
  return __int_as_float(i >= 0 ? i : (i ^ 0x7fffffff));
}

__global__ void gat_fill_i32(int* __restrict__ p, int v, long n) {
  long i = (long)blockIdx.x * blockDim.x + threadIdx.x;
  if (i < n) p[i] = v;
}

// pass 1: segment max over destinations (edges e >= E are self-loops (e-E, e-E))
__global__ void gat_edge_max(const int* __restrict__ src, const int* __restrict__ dst,
                             int E, int heads,
                             const float* __restrict__ as_in, const float* __restrict__ ad_in,
                             int* __restrict__ emax, long total) {
  long idx = (long)blockIdx.x * blockDim.x + threadIdx.x;
  if (idx >= total) return;
  long e = idx / heads;
  int hd = (int)(idx - e * heads);
  int s = (e < E) ? src[e] : (int)(e - E);
  int d = (e < E) ? dst[e] : (int)(e - E);
  float v = gat_leaky(as_in[(long)s * heads + hd] + ad_in[(long)d * heads + hd]);
  atomicMax(&emax[(long)d * heads + hd], gat_f2ord(v));
}

// pass 2: sum of exp(e - max) per destination segment
__global__ void gat_edge_sumexp(const int* __restrict__ src, const int* __restrict__ dst,
                                int E, int heads,
                                const float* __restrict__ as_in, const float* __restrict__ ad_in,
                                const int* __restrict__ emax, float* __restrict__ denom,
                                long total) {
  long idx = (long)blockIdx.x * blockDim.x + threadIdx.x;
  if (idx >= total) return;
  long e = idx / heads;
  int hd = (int)(idx - e * heads);
  int s = (e < E) ? src[e] : (int)(e - E);
  int d = (e < E) ? dst[e] : (int)(e - E);
  float v = gat_leaky(as_in[(long)s * heads + hd] + ad_in[(long)d * heads + hd]);
  float m = gat_ord2f(emax[(long)d * heads + hd]);
  atomicAdd(&denom[(long)d * heads + hd], expf(v - m));
}

// pass 3: normalized attention weight per (edge, head)
__global__ void gat_edge_alpha(const int* __restrict__ src, const int* __restrict__ dst,
                               int E, int heads,
                               const float* __restrict__ as_in, const float* __restrict__ ad_in,
                               const int* __restrict__ emax, const float* __restrict__ denom,
                               float* __restrict__ alpha, long total) {
  long idx = (long)blockIdx.x * blockDim.x + threadIdx.x;
  if (idx >= total) return;
  long e = idx / heads;
  int hd = (int)(idx - e * heads);
  int s = (e < E) ? src[e] : (int)(e - E);
  int d = (e < E) ? dst[e] : (int)(e - E);
  float v = gat_leaky(as_in[(long)s * heads + hd] + ad_in[(long)d * heads + hd]);
  float m = gat_ord2f(emax[(long)d * heads + hd]);
  float ex = expf(v - m);
  alpha[idx] = ex / (denom[(long)d * heads + hd] + 1e-16f);
}

// pass 4: scatter-add h[src]*alpha into out[dst]; one thread per (edge, channel)
__global__ void gat_edge_agg(const int* __restrict__ src, const int* __restrict__ dst,
                             int E, int heads, int F,
                             const float* __restrict__ h, const float* __restrict__ alpha,
                             float* __restrict__ out, long total) {
  long idx = (long)blockIdx.x * blockDim.x + threadIdx.x;
  if (idx >= total) return;
  const int HF = heads * F;
  long e = idx / HF;
  int c = (int)(idx - e * HF);
  int hd = c / F;
  int s = (e < E) ? src[e] : (int)(e - E);
  int d = (e < E) ? dst[e] : (int)(e - E);
  float al = alpha[e * heads + hd];
  atomicAdd(&out[(long)d * HF + c], h[(long)s * HF + c] * al);
}

__global__ void gat_bias_act(float* __restrict__ p, const float* __restrict__ bias,
                             long n, int C, int do_relu) {
  long i = (long)blockIdx.x * blockDim.x + threadIdx.x;
  if (i >= n) return;
  float v = p[i] + bias[i % C];
  p[i] = do_relu ? (v > 0.f ? v : 0.f) : v;
}

extern "C" void kernel_launch(void* const* d_in, const int* in_sizes, int n_in,
                              void* d_out, int out_size, void* d_ws, size_t ws_size,
                              hipStream_t stream) {
  const float* x      = (const float*)d_in[0];
  const int*   ei     = (const int*)d_in[1];
  const float* W1     = (const float*)d_in[2];
  const float* a_src1 = (const float*)d_in[3];
  const float* a_dst1 = (const float*)d_in[4];
  const float* b1     = (const float*)d_in[5];
  const float* W2     = (const float*)d_in[6];
  const float* a_src2 = (const float*)d_in[7];
  const float* a_dst2 = (const float*)d_in[8];
  const float* b2     = (const float*)d_in[9];
  float* out = (float*)d_out;

  const int* src = ei;
  const int* dst = ei + N_EDGES;
  const long Etot = (long)N_EDGES + N_NODES;   // self-loops appended logically
  const int HF1 = HEADS * HID;                 // 128

  // workspace layout (floats); h2 reuses h1's region
  float* ws    = (float*)d_ws;
  float* h1    = ws;                                     // N*128
  float* agg1  = h1   + (long)N_NODES * HF1;             // N*128
  float* asb   = agg1 + (long)N_NODES * HF1;             // N*HEADS
  float* adb   = asb  + (long)N_NODES * HEADS;           // N*HEADS
  int*   emax  = (int*)(adb + (long)N_NODES * HEADS);    // N*HEADS ints
  float* denom = (float*)(emax + (long)N_NODES * HEADS); // N*HEADS
  float* alpha = denom + (long)N_NODES * HEADS;          // Etot*HEADS
  float* h2    = h1;                                     // N*64 (reuse)

  const int TB = 256;
  const long nh1 = (long)N_NODES * HEADS;
  const long eh1 = Etot * HEADS;
  const long ec1 = Etot * HF1;

  // ---------------- layer 1: heads=4, F=32, concat ----------------
  gat_gemm_wmma_tdm<IN_DIM, HEADS * HID>
      <<<N_NODES / 16, 32 * (HF1 / 16), 0, stream>>>(x, W1, h1);
  gat_attn_coef<<<(int)((nh1 + TB - 1) / TB), TB, 0, stream>>>(h1, a_src1, a_dst1, asb, adb,
                                                               N_NODES, HEADS, HID);
  gat_fill_i32<<<(int)((nh1 + TB - 1) / TB), TB, 0, stream>>>(emax, (int)0x80000000, nh1);
  (void)hipMemsetAsync(denom, 0, nh1 * sizeof(float), stream);
  (void)hipMemsetAsync(agg1, 0, (long)N_NODES * HF1 * sizeof(float), stream);

  gat_edge_max<<<(int)((eh1 + TB - 1) / TB), TB, 0, stream>>>(src, dst, N_EDGES, HEADS,
                                                              asb, adb, emax, eh1);
  gat_edge_sumexp<<<(int)((eh1 + TB - 1) / TB), TB, 0, stream>>>(src, dst, N_EDGES, HEADS,
                                                                 asb, adb, emax, denom, eh1);
  gat_edge_alpha<<<(int)((eh1 + TB - 1) / TB), TB, 0, stream>>>(src, dst, N_EDGES, HEADS,
                                                                asb, adb, emax, denom, alpha, eh1);
  gat_edge_agg<<<(int)((ec1 + TB - 1) / TB), TB, 0, stream>>>(src, dst, N_EDGES, HEADS, HID,
                                                              h1, alpha, agg1, ec1);
  gat_bias_act<<<(int)(((long)N_NODES * HF1 + TB - 1) / TB), TB, 0, stream>>>(
      agg1, b1, (long)N_NODES * HF1, HF1, /*relu=*/1);

  // ---------------- layer 2: heads=1, F=64, mean(=identity) ----------------
  const long nh2 = N_NODES;
  const long eh2 = Etot;
  const long ec2 = Etot * OUT_DIM;

  gat_gemm_wmma_tdm<HEADS * HID, OUT_DIM>
      <<<N_NODES / 16, 32 * (OUT_DIM / 16), 0, stream>>>(agg1, W2, h2);
  gat_attn_coef<<<(int)((nh2 + TB - 1) / TB), TB, 0, stream>>>(h2, a_src2, a_dst2, asb, adb,
                                                               N_NODES, 1, OUT_DIM);
  gat_fill_i32<<<(int)((nh2 + TB - 1) / TB), TB, 0, stream>>>(emax, (int)0x80000000, nh2);
  (void)hipMemsetAsync(denom, 0, nh2 * sizeof(float), stream);
  (void)hipMemsetAsync(out, 0, (long)N_NODES * OUT_DIM * sizeof(float), stream);

  gat_edge_max<<<(int)((eh2 + TB - 1) / TB), TB, 0, stream>>>(src, dst, N_EDGES, 1,
                                                              asb, adb, emax, eh2);
  gat_edge_sumexp<<<(int)((eh2 + TB - 1) / TB), TB, 0, stream>>>(src, dst, N_EDGES, 1,
                                                                 asb, adb, emax, denom, eh2);
  gat_edge_alpha<<<(int)((eh2 + TB - 1) / TB), TB, 0, stream>>>(src, dst, N_EDGES, 1,
                                                                asb, adb, emax, denom, alpha, eh2);
  gat_edge_agg<<<(int)((ec2 + TB - 1) / TB), TB, 0, stream>>>(src, dst, N_EDGES, 1, OUT_DIM,
                                                              h2, alpha, out, ec2);
  gat_bias_act<<<(int)(((long)N_NODES * OUT_DIM + TB - 1) / TB), TB, 0, stream>>>(
      out, b2, (long)N_NODES * OUT_DIM, OUT_DIM, /*relu=*/0);
}